// GCN_20495583937193
// MI455X (gfx1250) — compile-verified
//
#include <hip/hip_runtime.h>
#include <hip/hip_bf16.h>

// ---------------------------------------------------------------------------
// 2-layer GCN for MI455X (gfx1250, wave32).
//  - Dense GEMMs (N x 128 @ 128 x 128) via V_WMMA_F32_16X16X4_F32 (fp32 exact).
//  - Edge aggregation (the real bottleneck, ~L2-resident) via float4 gathers
//    + global f32 atomic adds, one wave per edge (self-loops fused in).
// ---------------------------------------------------------------------------

typedef float v2f __attribute__((ext_vector_type(2)));
typedef float v8f __attribute__((ext_vector_type(8)));

#define D 128           // feature dim (d_in == d_hid == 128)
#define XS_STRIDE 132   // padded LDS row stride (floats) to avoid bank conflicts

// ---------------- small utility kernels ----------------

__global__ void fill_f32(float* __restrict__ p, float v, int n) {
    int i = blockIdx.x * blockDim.x + threadIdx.x;
    if (i < n) p[i] = v;
}

__global__ void deg_accum(const int* __restrict__ dst, float* __restrict__ deg, int nE) {
    int e = blockIdx.x * blockDim.x + threadIdx.x;
    if (e < nE) atomicAdd(&deg[dst[e]], 1.0f);
}

__global__ void deg_to_dinv(float* __restrict__ d, int n) {
    int i = blockIdx.x * blockDim.x + threadIdx.x;
    if (i < n) d[i] = rsqrtf(d[i]);   // deg >= 1 always (self-loop)
}

__global__ void bias_relu(float* __restrict__ h, const float* __restrict__ b, int n) {
    int i = blockIdx.x * blockDim.x + threadIdx.x;
    if (i < n) {
        float v = h[i] + b[i & (D - 1)];
        h[i] = v > 0.0f ? v : 0.0f;
    }
}

__global__ void bias_add(float* __restrict__ h, const float* __restrict__ b, int n) {
    int i = blockIdx.x * blockDim.x + threadIdx.x;
    if (i < n) h[i] = h[i] + b[i & (D - 1)];
}

// ---------------- fp32 WMMA GEMM: H[N x 128] = X[N x 128] @ W[128 x 128] ----
// Block = 256 threads = 8 waves. Block covers 16 output rows; wave w owns the
// 16x16 tile at columns [16w, 16w+16). X strip staged in LDS; W streamed from
// global (64 KB, L2-hot across all 6250 blocks).
__global__ void gemm_wmma_f32(const float* __restrict__ X,
                              const float* __restrict__ W,
                              float* __restrict__ H, int nRows) {
    __shared__ float xs[16 * XS_STRIDE];

    const int row0 = blockIdx.x * 16;
    const int tid  = threadIdx.x;

    // Stage 16 x 128 strip of X into LDS (512 float4 loads across 256 threads).
    for (int i = tid; i < 16 * (D / 4); i += 256) {
        const int r = i >> 5;        // 32 float4 per row
        const int c = i & 31;
        float4 v = make_float4(0.f, 0.f, 0.f, 0.f);
        if (row0 + r < nRows)
            v = ((const float4*)(X + (size_t)(row0 + r) * D))[c];
        float* p = &xs[r * XS_STRIDE + c * 4];
        p[0] = v.x; p[1] = v.y; p[2] = v.z; p[3] = v.w;
    }
    __syncthreads();

    const int wave = tid >> 5;           // 0..7 -> column tile
    const int lane = tid & 31;
    const int m    = lane & 15;          // A-matrix row (lanes 0-15 / 16-31 both M=0..15)
    const int kk   = (lane >> 4) << 1;   // K sub-pair: 0 (lo lanes) or 2 (hi lanes)
    const int n    = lane & 15;          // B/C/D column within tile
    const int col  = wave * 16 + n;      // global output column

    v8f acc = {};
    #pragma unroll
    for (int k0 = 0; k0 < D; k0 += 4) {
        v2f a, b;
        // A 16x4 fragment per ISA layout: VGPR0=K+kk, VGPR1=K+kk+1
        a.x = xs[m * XS_STRIDE + k0 + kk];
        a.y = xs[m * XS_STRIDE + k0 + kk + 1];
        // B 4x16 fragment: row K+kk striped across lanes (mirrors A's K split)
        b.x = W[(size_t)(k0 + kk) * D + col];
        b.y = W[(size_t)(k0 + kk + 1) * D + col];
        acc = __builtin_amdgcn_wmma_f32_16x16x4_f32(
            /*neg_a=*/false, a, /*neg_b=*/false, b,
            /*c_mod=*/(short)0, acc, /*reuse_a=*/false, /*reuse_b=*/false);
    }

    // D layout: VGPR r holds row (r + 8*lane[4]) of the tile, column n.
    const int halfm = (lane >> 4) * 8;
    #pragma unroll
    for (int r = 0; r < 8; ++r) {
        const int row = row0 + halfm + r;
        if (row < nRows) H[(size_t)row * D + col] = acc[r];
    }
}

// ---------------- edge aggregation: out[dst] += h[src] * dinv[src]*dinv[dst] -
// One wave per edge. Edges [0,nE) come from edge_index; edges [nE, nE+nN) are
// the implicit self-loops. Each lane handles 4 consecutive floats (float4 load
// -> global_load_b128, then 4x global_atomic_add_f32).
__global__ void edge_aggregate(const int* __restrict__ src,
                               const int* __restrict__ dst,
                               const float* __restrict__ dinv,
                               const float* __restrict__ h,
                               float* __restrict__ out, int nE, int nN) {
    const int warp = (blockIdx.x * blockDim.x + threadIdx.x) >> 5;
    const int lane = threadIdx.x & 31;
    const int total = nE + nN;
    if (warp >= total) return;

    int s, d;
    if (warp < nE) { s = src[warp]; d = dst[warp]; }
    else           { s = d = warp - nE; }          // self-loop

    const float norm = dinv[s] * dinv[d];
    const float4 v = ((const float4*)(h + (size_t)s * D))[lane];

    float* o = out + (size_t)d * D + lane * 4;
    atomicAdd(o + 0, v.x * norm);
    atomicAdd(o + 1, v.y * norm);
    atomicAdd(o + 2, v.z * norm);
    atomicAdd(o + 3, v.w * norm);
}

// ---------------------------------------------------------------------------

static inline size_t align256(size_t v) { return (v + 255) & ~(size_t)255; }

extern "C" void kernel_launch(void* const* d_in, const int* in_sizes, int n_in,
                              void* d_out, int out_size, void* d_ws, size_t ws_size,
                              hipStream_t stream) {
    const float* x  = (const float*)d_in[0];
    const int*   ei = (const int*)d_in[1];   // edge_index [2, E] (int32)
    const float* W1 = (const float*)d_in[2];
    const float* b1 = (const float*)d_in[3];
    const float* W2 = (const float*)d_in[4];
    const float* b2 = (const float*)d_in[5];
    float* out = (float*)d_out;

    const int N = in_sizes[0] / D;
    const int E = in_sizes[1] / 2;
    const int* srcIdx = ei;         // edge_index[0]
    const int* dstIdx = ei + E;     // edge_index[1]

    // Workspace layout: dinv[N] | hbuf[N*128] | agg[N*128]  (~103 MB)
    char* ws = (char*)d_ws;
    float* dinv = (float*)ws;
    const size_t o1 = align256((size_t)N * sizeof(float));
    float* hbuf = (float*)(ws + o1);
    const size_t o2 = o1 + align256((size_t)N * D * sizeof(float));
    float* agg  = (float*)(ws + o2);

    const int nF = N * D;
    const int blk = 256;
    const int gN   = (N + blk - 1) / blk;
    const int gE   = (E + blk - 1) / blk;
    const int gF   = (nF + blk - 1) / blk;
    const int gGemm = (N + 15) / 16;
    const int gEdge = ((E + N) + 7) / 8;   // 8 waves per block, 1 wave per edge

    // --- degree / norm (self-loop contributes the initial 1.0) ---
    fill_f32   <<<gN, blk, 0, stream>>>(dinv, 1.0f, N);
    deg_accum  <<<gE, blk, 0, stream>>>(dstIdx, dinv, E);
    deg_to_dinv<<<gN, blk, 0, stream>>>(dinv, N);

    // --- layer 1: h = relu(A_norm (x @ W1) + b1) ---
    gemm_wmma_f32 <<<gGemm, blk, 0, stream>>>(x, W1, hbuf, N);
    fill_f32      <<<gF, blk, 0, stream>>>(agg, 0.0f, nF);
    edge_aggregate<<<gEdge, blk, 0, stream>>>(srcIdx, dstIdx, dinv, hbuf, agg, E, N);
    bias_relu     <<<gF, blk, 0, stream>>>(agg, b1, nF);

    // --- layer 2: out = A_norm (h @ W2) + b2 ---
    gemm_wmma_f32 <<<gGemm, blk, 0, stream>>>(agg, W2, hbuf, N);
    fill_f32      <<<gF, blk, 0, stream>>>(out, 0.0f, nF);
    edge_aggregate<<<gEdge, blk, 0, stream>>>(srcIdx, dstIdx, dinv, hbuf, out, E, N);
    bias_add      <<<gF, blk, 0, stream>>>(out, b2, nF);
}